// CIE_18236431138961
// MI455X (gfx1250) — compile-verified
//
#include <hip/hip_runtime.h>

// out[n,h,d] = sum_s x[n,s,d] * W[s,h],  W[s,h] = FM[s+1,h] * Agg[0,s,h]
// (exact algebraic collapse of the reference's sort/gap/code Choquet form,
//  since the lookup table is additive over set bits)

typedef __attribute__((ext_vector_type(2))) float v2f;
typedef __attribute__((ext_vector_type(4))) float v4f;
typedef __attribute__((ext_vector_type(8))) float v8f;

#define SRC 16   // S
#define HD  4    // H
#define NB  128  // N
#define DD  512  // D

__global__ __launch_bounds__(256)
void CIE_choquet_wmma(const float* __restrict__ x,
                      const float* __restrict__ FM,
                      const float* __restrict__ Agg,
                      float* __restrict__ out)
{
    const int lane = threadIdx.x & 31;
    const int wave = threadIdx.x >> 5;
    const int tile = blockIdx.x * 8 + wave;   // 4096 tiles total
    const int n    = tile >> 5;               // tile / (D/16)
    const int d0   = (tile & 31) << 4;        // 16 d-values per tile

    const int half = lane >> 4;               // 0: lanes 0-15, 1: lanes 16-31
    const int lm   = lane & 15;               // A: M (d offset); B/D: N (h)

    // ---- B fragments: W[s,h] padded to 16 columns (h>=4 -> 0), branch-free
    // Lane layout (mirrors 32-bit A 16x4): v0 = B[K=4k+2*half,  N=lm]
    //                                      v1 = B[K=4k+2*half+1,N=lm]
    const float msk = (lm < HD) ? 1.0f : 0.0f;
    const int   h   = lm & (HD - 1);          // clamped, always in-bounds
    v2f b[4];
#pragma unroll
    for (int k = 0; k < 4; ++k) {
        const int s0 = 4 * k + 2 * half;
        b[k].x = msk * FM[(s0 + 1) * HD + h] * Agg[ s0      * HD + h];
        b[k].y = msk * FM[(s0 + 2) * HD + h] * Agg[(s0 + 1) * HD + h];
    }

    // ---- A fragments + chained K=4 f32 WMMAs over s = 0..15
    // A[M=d, K=s] = x[n, s, d0+lm]; per ISA: lanes 0-15 K={0,1}, 16-31 K={2,3}
    const float* xb = x + (size_t)n * SRC * DD + (size_t)d0 + lm;
    v8f c = {};
#pragma unroll
    for (int k = 0; k < 4; ++k) {
        const int s0 = 4 * k + 2 * half;
        v2f a;
        a.x = xb[(size_t)(s0)     * DD];
        a.y = xb[(size_t)(s0 + 1) * DD];
        c = __builtin_amdgcn_wmma_f32_16x16x4_f32(
                /*neg_a=*/false, a, /*neg_b=*/false, b[k],
                /*c_mod=*/(short)0, c, /*reuse_a=*/false, /*reuse_b=*/false);
    }

    // ---- Store: lane (half,lm=h<4) holds out[d0 + 8*half + r, h] in c[r],
    // which is 8 contiguous d's in out (N,H,D) row-major.
    if (lm < HD) {
        float* op = out + (size_t)n * HD * DD + (size_t)lm * DD + d0 + 8 * half;
        v4f lo = { c[0], c[1], c[2], c[3] };
        v4f hi = { c[4], c[5], c[6], c[7] };
        *(v4f*)(op)     = lo;
        *(v4f*)(op + 4) = hi;
    }
}

extern "C" void kernel_launch(void* const* d_in, const int* in_sizes, int n_in,
                              void* d_out, int out_size, void* d_ws, size_t ws_size,
                              hipStream_t stream) {
    (void)in_sizes; (void)n_in; (void)out_size; (void)d_ws; (void)ws_size;
    const float* x   = (const float*)d_in[0];  // (N,S,D) f32
    const float* FM  = (const float*)d_in[1];  // (S+1,H) f32
    const float* Agg = (const float*)d_in[2];  // (1,S,H) f32
    // d_in[3] (source_index) unused: folded into W analytically.
    float* out = (float*)d_out;                // (N,H,D) f32

    const int tiles  = NB * (DD / 16);         // 4096
    const int blocks = tiles / 8;              // 8 waves (256 thr) per block
    CIE_choquet_wmma<<<blocks, 256, 0, stream>>>(x, FM, Agg, out);
}